// kNNModule_41248865911197
// MI455X (gfx1250) — compile-verified
//
#include <hip/hip_runtime.h>
#include <stdint.h>

typedef float v2f __attribute__((ext_vector_type(2)));
typedef float v8f __attribute__((ext_vector_type(8)));

#define KNN 4
#define MQ 4000
#define NP 50000

__global__ __launch_bounds__(32)
void knn_wmma_kernel(const float* __restrict__ points,
                     const float* __restrict__ queries,
                     float* __restrict__ out)
{
    // one wave per block: LDS traffic is intra-wave => in-order, no barriers.
    // 16x16 d^2 tile; 20-col pitch keeps 8-float slices 16B aligned
    __shared__ __align__(16) float tile[16][20];
    __shared__ int64_t mk[32][KNN];   // per-lane partial top-4 packed keys

    const int lane = threadIdx.x & 31;
    const int t    = blockIdx.x;                // query tile id, 0..249
    const int m0   = t * 16;
    const int half = lane >> 4;                 // 0: lanes 0-15, 1: lanes 16-31
    const int l15  = lane & 15;
    const int col0 = 8 * half;                  // this lane's column range base

    // ---- A = -2*Q (16x4, K padded). ISA 32-bit A 16x4 layout:
    // lanes 0-15: VGPR0=K0, VGPR1=K1; lanes 16-31: VGPR0=K2, VGPR1=K3
    v2f a;
    {
        const int qrow = m0 + l15;
        const float qx = queries[3 * qrow + 0];
        const float qy = queries[3 * qrow + 1];
        const float qz = queries[3 * qrow + 2];
        a.x = -2.0f * (half ? qz : qx);
        a.y = half ? 0.0f : (-2.0f * qy);
    }

    // ---- |q|^2 for the 8 accumulator rows this lane covers (C row = i + 8*half)
    float q2r[8];
#pragma unroll
    for (int i = 0; i < 8; ++i) {
        const int qrow = m0 + i + 8 * half;
        const float qx = queries[3 * qrow + 0];
        const float qy = queries[3 * qrow + 1];
        const float qz = queries[3 * qrow + 2];
        q2r[i] = qx * qx + qy * qy + qz * qz;
    }

    // ---- running top-4 as packed keys (d2_bits<<32 | idx), ascending by
    // SIGNED 64-bit compare: non-negative floats order as ints; a rounding-
    // negative d2 (reference clamps it to 0 = best) sorts first. Low word
    // gives lower-index-first tie-breaking (== stable top_k).
    int64_t K[KNN];
#pragma unroll
    for (int k = 0; k < KNN; ++k) K[k] = 0x7fffffffffffffffLL;
    float bd3f = 3.4e38f;   // shadow of current 4th-best d^2 for the cheap filter

    // ---- software pipeline: hold current tile's point in regs, load next early
    float px = points[3 * l15 + 0];
    float py = points[3 * l15 + 1];
    float pz = points[3 * l15 + 2];

    for (int base = 0; base < NP; base += 16) {
        float nx = 0.0f, ny = 0.0f, nz = 0.0f;
        if (base + 16 < NP) {
            const int nn = base + 16 + l15;
            nx = points[3 * nn + 0];
            ny = points[3 * nn + 1];
            nz = points[3 * nn + 2];
            __builtin_prefetch(&points[3 * (nn + 16)], 0, 3);  // 2 tiles ahead
        }

        const float p2 = px * px + py * py + pz * pz;

        // ---- B (4x16): mirrored striping of A
        v2f b;
        b.x = half ? pz : px;
        b.y = half ? 0.0f : py;

        // ---- seed accumulator with q^2 + p^2, then D = C + A*B == d^2 tile
        v8f c;
#pragma unroll
        for (int i = 0; i < 8; ++i) c[i] = q2r[i] + p2;

        c = __builtin_amdgcn_wmma_f32_16x16x4_f32(
                /*neg_a=*/false, a, /*neg_b=*/false, b,
                /*c_mod=*/(short)0, c, /*reuse_a=*/false, /*reuse_b=*/false);

        // stage d^2 tile in LDS (unclamped): row = i + 8*half, col = l15
#pragma unroll
        for (int i = 0; i < 8; ++i) tile[i + 8 * half][l15] = c[i];

        // every lane owns (row = l15, cols col0..col0+7): 2x ds_load_b128
        // (same wave wrote it; DS ops are in-order per wave, no barrier needed)
        const float4 t0 = *(const float4*)&tile[l15][col0];
        const float4 t1 = *(const float4*)&tile[l15][col0 + 4];
        const float cand[8] = {t0.x, t0.y, t0.z, t0.w, t1.x, t1.y, t1.z, t1.w};

        // cheap filter: min of 8 candidates vs current 4th-best (rarely passes)
        const float mn = fminf(fminf(fminf(cand[0], cand[1]), fminf(cand[2], cand[3])),
                               fminf(fminf(cand[4], cand[5]), fminf(cand[6], cand[7])));
        if (mn < bd3f) {
#pragma unroll
            for (int j = 0; j < 8; ++j) {
                const int64_t key = (int64_t)(
                    ((uint64_t)__float_as_uint(cand[j]) << 32) |
                    (uint64_t)(uint32_t)(base + col0 + j));
                if (key < K[KNN - 1]) {
                    K[KNN - 1] = key;
#pragma unroll
                    for (int k = KNN - 1; k > 0; --k) {
                        if (K[k] < K[k - 1]) {
                            int64_t tk = K[k]; K[k] = K[k - 1]; K[k - 1] = tk;
                        }
                    }
                }
            }
            bd3f = __uint_as_float((uint32_t)((uint64_t)K[KNN - 1] >> 32));
        }

        px = nx; py = ny; pz = nz;   // rotate pipeline regs
    }

    // ---- publish partial lists, lanes 0-15 merge own half with lane+16
#pragma unroll
    for (int k = 0; k < KNN; ++k) mk[lane][k] = K[k];

    if (half == 0) {
        const int row = m0 + l15;
        int ia = 0, ib = 0;
#pragma unroll
        for (int k = 0; k < KNN; ++k) {
            const int64_t ka = mk[l15][ia];
            const int64_t kb = mk[l15 + 16][ib];
            const bool takeA = ka < kb;         // (d^2, idx) lexicographic
            const int64_t key = takeA ? ka : kb;
            ia += takeA ? 1 : 0;
            ib += takeA ? 0 : 1;
            const float d2 = __uint_as_float((uint32_t)((uint64_t)key >> 32));
            out[row * KNN + k]            = sqrtf(fmaxf(d2, 0.0f));
            out[MQ * KNN + row * KNN + k] = (float)(uint32_t)((uint64_t)key & 0xffffffffu);
        }
    }
}

extern "C" void kernel_launch(void* const* d_in, const int* in_sizes, int n_in,
                              void* d_out, int out_size, void* d_ws, size_t ws_size,
                              hipStream_t stream) {
    const float* points  = (const float*)d_in[0];   // [50000, 3]
    const float* queries = (const float*)d_in[1];   // [4000, 3]
    float* out = (float*)d_out;                     // [4000*4 dist | 4000*4 idx]

    // 250 query tiles of 16 rows; 1 wave per block -> no barriers at all
    dim3 grid(250), block(32);
    knn_wmma_kernel<<<grid, block, 0, stream>>>(points, queries, out);
}